// LiftSplatShootTransformer_25572235280719
// MI455X (gfx1250) — compile-verified
//
#include <hip/hip_runtime.h>
#include <hip/hip_bf16.h>

typedef float v2f __attribute__((ext_vector_type(2)));
typedef float v8f __attribute__((ext_vector_type(8)));

// Problem constants
#define BB 32
#define NN 6
#define DD 41
#define CC 64
#define FH 4
#define FW 6
#define INCH 768
#define NXY 200
#define PP   (BB*NN*FH*FW)        // 4608 spatial points
#define NPTS (BB*NN*DD*FH*FW)     // 188928 frustum points
#define YLD  112                  // padded rows of y per point (105 -> 112)
#define POOL_ELEMS 81920000LL     // 32*64*200*200

// ---------------- zero the pooled/final output (328 MB, float4 stores) ----------------
__global__ __launch_bounds__(256) void k_zero(float4* __restrict__ out, int n4) {
  int i = blockIdx.x * 256 + threadIdx.x;
  if (i < n4) out[i] = make_float4(0.f, 0.f, 0.f, 0.f);
}

// ---------------- combine[bn] = rots[bn] @ inv(intrins[bn])  (192 matrices) ----------
__global__ void k_combine(const float* __restrict__ rots,
                          const float* __restrict__ intrins,
                          float* __restrict__ comb) {
  int bn = blockIdx.x * blockDim.x + threadIdx.x;
  if (bn >= BB * NN) return;
  const float* K = intrins + bn * 9;
  float a=K[0],b=K[1],c=K[2],d=K[3],e=K[4],f=K[5],g=K[6],h=K[7],i=K[8];
  float A =  (e*i - f*h), Bv = -(b*i - c*h), Cv =  (b*f - c*e);
  float Dv = -(d*i - f*g), E =  (a*i - c*g), F = -(a*f - c*d);
  float G =  (d*h - e*g), H = -(a*h - b*g), I =  (a*e - b*d);
  float det = a*A + b*Dv + c*G;
  float s = 1.0f / det;
  float M[9] = {A*s, Bv*s, Cv*s, Dv*s, E*s, F*s, G*s, H*s, I*s};
  const float* R = rots + bn * 9;
  float* O = comb + bn * 9;
  #pragma unroll
  for (int r = 0; r < 3; ++r)
    #pragma unroll
    for (int cc = 0; cc < 3; ++cc)
      O[r*3+cc] = R[r*3+0]*M[0*3+cc] + R[r*3+1]*M[1*3+cc] + R[r*3+2]*M[2*3+cc];
}

// ---------------- y[p,o] = sum_c w[o,c]*x[bn,c,hw] + bias[o]  via fp32 WMMA ----------
// One wave32 per 16x16 output tile. M tiles: 7 (105->112). N tiles: 288 (4608).
// 2016 tiles = 252 blocks * 8 waves exactly.
__global__ __launch_bounds__(256) void k_gemm(const float* __restrict__ x,
                                              const float* __restrict__ w,
                                              const float* __restrict__ bias,
                                              float* __restrict__ y) {
  const int lane = threadIdx.x & 31;
  const int tile = blockIdx.x * 8 + (threadIdx.x >> 5);   // 0..2015
  const int mT   = tile % 7;
  const int nT   = tile / 7;
  const int half = lane >> 4;       // 0 or 1
  const int l16  = lane & 15;
  const int m    = mT * 16 + l16;   // output-channel row this lane loads for A
  const int p    = nT * 16 + l16;   // spatial column this lane loads for B
  const int bn   = p / (FH * FW);
  const int hw   = p % (FH * FW);
  const float* xcol = x + (size_t)bn * (INCH * FH * FW) + hw;   // + c*24
  const int   mC  = (m < 105) ? m : 104;
  const float msk = (m < 105) ? 1.0f : 0.0f;
  const float* wrow = w + (size_t)mC * INCH;

  v8f acc = {};
  #pragma unroll 4
  for (int k = 0; k < INCH; k += 4) {
    const int kk = k + 2 * half;
    // A fragment: A[m][kk], A[m][kk+1]  (contiguous, 8B aligned)
    v2f a = *(const v2f*)(wrow + kk);
    a.x *= msk; a.y *= msk;
    // B fragment: B[kk][p], B[kk+1][p]  (stride 24 floats in x)
    v2f b;
    b.x = xcol[(kk + 0) * (FH * FW)];
    b.y = xcol[(kk + 1) * (FH * FW)];
    acc = __builtin_amdgcn_wmma_f32_16x16x4_f32(false, a, false, b,
                                                (short)0, acc, false, false);
  }

  // D layout: VGPR r -> row o = mT*16 + 8*half + r, col = p
  const int obase = mT * 16 + 8 * half;
  float* yp = y + (size_t)p * YLD;
  #pragma unroll
  for (int r = 0; r < 8; ++r) {
    int o = obase + r;
    if (o < 105) yp[o] = acc[r] + bias[o];
  }
}

// ---------------- per-point depth softmax (41 logits contiguous in y_ws) -------------
// Also streams raw logits to the depth_logit output region.
__global__ __launch_bounds__(256) void k_softmax(float* __restrict__ y,
                                                 float* __restrict__ dlog) {
  int p = blockIdx.x * 256 + threadIdx.x;     // 0..4607, exact
  float* col = y + (size_t)p * YLD;
  int bn = p / (FH * FW), hw = p % (FH * FW);
  float* dl = dlog + (size_t)bn * (DD * FH * FW) + hw;
  float v[DD];
  float mx = -3.4e38f;
  #pragma unroll
  for (int d = 0; d < DD; ++d) {
    v[d] = col[d];
    dl[d * (FH * FW)] = v[d];                 // raw depth logit output
    mx = fmaxf(mx, v[d]);
  }
  float s = 0.f;
  #pragma unroll
  for (int d = 0; d < DD; ++d) { float e = expf(v[d] - mx); v[d] = e; s += e; }
  float is = 1.0f / s;
  #pragma unroll
  for (int d = 0; d < DD; ++d) col[d] = v[d] * is;
}

// ---------------- voxel scatter: 64 threads per frustum point, atomic f32 adds -------
__global__ __launch_bounds__(256) void k_scatter(const float* __restrict__ y,
                                                 const float* __restrict__ comb,
                                                 const float* __restrict__ trans,
                                                 float* __restrict__ out) {
  const int t   = blockIdx.x * 256 + threadIdx.x;
  const int pt  = t >> 6;                     // frustum point id, exact fit
  const int c   = t & 63;                     // channel
  const int bn  = pt / (DD * FH * FW);
  const int rem = pt % (DD * FH * FW);
  const int d   = rem / (FH * FW);
  const int hw  = rem % (FH * FW);
  const int h   = hw / FW, wv = hw % FW;
  const int b   = bn / NN;

  // frustum constants (np.linspace values rounded to fp32)
  const float XS[FW] = {0.f, 39.8f, 79.6f, 119.4f, 159.2f, 199.f};
  const float YS[FH] = {0.f, 49.666668f, 99.333336f, 149.f};
  const float dep = 4.0f + (float)d;
  const float px = XS[wv] * dep;
  const float py = YS[h]  * dep;
  const float pz = dep;

  const float* M = comb + bn * 9;
  const float* T = trans + bn * 3;
  float g0 = M[0]*px + M[1]*py + M[2]*pz + T[0];
  float g1 = M[3]*px + M[4]*py + M[5]*pz + T[1];
  float g2 = M[6]*px + M[7]*py + M[8]*pz + T[2];

  // (geom - (BX - DX/2)) / DX, truncate toward zero like .astype(int32)
  int gx = (int)((g0 + 50.0f) * 2.0f);        // /0.5 exact
  int gy = (int)((g1 + 50.0f) * 2.0f);
  int gz = (int)((g2 + 10.0f) / 20.0f);
  bool kept = (gx >= 0) & (gx < NXY) & (gy >= 0) & (gy < NXY) & (gz == 0);

  const int p = bn * (FH * FW) + hw;
  float val = y[(size_t)p * YLD + d] * y[(size_t)p * YLD + DD + c];
  if (kept) {
    // final[b, c, gx, gy]
    atomicAdd(out + ((size_t)(b * CC + c) * (NXY * NXY) + gx * NXY + gy), val);
  }
}

extern "C" void kernel_launch(void* const* d_in, const int* in_sizes, int n_in,
                              void* d_out, int out_size, void* d_ws, size_t ws_size,
                              hipStream_t stream) {
  const float* x       = (const float*)d_in[0];
  const float* rots    = (const float*)d_in[1];
  const float* trans   = (const float*)d_in[2];
  const float* intrins = (const float*)d_in[3];
  const float* conv_w  = (const float*)d_in[4];
  const float* conv_b  = (const float*)d_in[5];
  float* out = (float*)d_out;

  float* y_ws = (float*)d_ws;                         // 4608*112 floats (~2 MB)
  float* comb = y_ws + (size_t)PP * YLD;              // 192*9 floats

  // 1) zero final output (81,920,000 floats = 20,480,000 float4)
  k_zero<<<80000, 256, 0, stream>>>((float4*)out, 20480000);
  // 2) combine matrices
  k_combine<<<1, 192, 0, stream>>>(rots, intrins, comb);
  // 3) fp32 WMMA GEMM -> y_ws
  k_gemm<<<252, 256, 0, stream>>>(x, conv_w, conv_b, y_ws);
  // 4) softmax over depth + emit raw logits to d_out tail
  k_softmax<<<PP / 256, 256, 0, stream>>>(y_ws, out + POOL_ELEMS);
  // 5) voxel-pool scatter into final layout
  k_scatter<<<(NPTS * 64) / 256, 256, 0, stream>>>(y_ws, comb, trans, out);
}